// ToRGBLayer_13932873909236
// MI455X (gfx1250) — compile-verified
//
#include <hip/hip_runtime.h>

typedef __attribute__((ext_vector_type(2))) float v2f;
typedef __attribute__((ext_vector_type(4))) float f4;
typedef __attribute__((ext_vector_type(8))) float v8f;

#define B_    16
#define CIN_  128
#define COUT_ 3
#define WDIM_ 512
#define HW_   65536

// ---------------------------------------------------------------------------
// Kernel 1: style = w @ affine_w^T + affine_b   (16x128, K=512) via
// V_WMMA_F32_16X16X4_F32, then fuse into the 1x1-conv weight:
//   eff[b][o][i] = style[b][i] * weight[o][i]      -> workspace (16*3*128 f32)
// One workgroup, 8 waves; wave j owns CIN columns [16j, 16j+16).
// ---------------------------------------------------------------------------
__global__ __launch_bounds__(256) void style_eff_kernel(
    const float* __restrict__ w,        // (16,512)
    const float* __restrict__ weight,   // (3,128)  (COUT,CIN)
    const float* __restrict__ affine_w, // (128,512)
    const float* __restrict__ affine_b, // (128)
    float* __restrict__ eff)            // ws: (16,3,128)
{
  __shared__ float style_lds[B_ * CIN_];

  const int tid  = threadIdx.x;
  const int wave = tid >> 5;     // 0..7 -> 16-wide CIN tile
  const int lane = tid & 31;
  const int half = lane >> 4;    // K sub-pair select per ISA A/B layout
  const int ln   = lane & 15;

  const int cidx = wave * 16 + ln;   // CIN column this lane supplies for B

  v8f c = {0.f, 0.f, 0.f, 0.f, 0.f, 0.f, 0.f, 0.f};
  for (int k = 0; k < WDIM_; k += 4) {
    const int kk = k + 2 * half;
    // A 16x4 (M=batch rows): VGPR0=K(kk), VGPR1=K(kk+1)
    v2f a;
    a.x = w[ln * WDIM_ + kk];
    a.y = w[ln * WDIM_ + kk + 1];
    // B 4x16 = affine_w^T tile: B[k][n] = affine_w[cidx][k]
    v2f bm;
    bm.x = affine_w[cidx * WDIM_ + kk];
    bm.y = affine_w[cidx * WDIM_ + kk + 1];
    c = __builtin_amdgcn_wmma_f32_16x16x4_f32(
        /*neg_a=*/false, a, /*neg_b=*/false, bm,
        /*c_mod=*/(short)0, c, /*reuse_a=*/false, /*reuse_b=*/false);
  }

  // C/D layout: element r -> (M=r, N=ln) for lanes 0-15, (M=r+8, N=ln) for 16-31
#pragma unroll
  for (int r = 0; r < 8; ++r) {
    const int bi = half ? (r + 8) : r;
    style_lds[bi * CIN_ + cidx] = c[r] + affine_b[cidx];
  }
  __syncthreads();

  // Fold style into the conv weight (tiny: 6144 values).
  for (int idx = tid; idx < B_ * COUT_ * CIN_; idx += 256) {
    const int b   = idx / (COUT_ * CIN_);
    const int rem = idx - b * (COUT_ * CIN_);
    const int o   = rem / CIN_;
    const int i   = rem - o * CIN_;
    eff[idx] = style_lds[b * CIN_ + i] * weight[o * CIN_ + i];
  }
}

// ---------------------------------------------------------------------------
// Kernel 2: HBM-bound streaming pass.
//   out[b][o][pix] = sum_i x[b][i][pix] * eff[b][o][i] + bias[o]
// Grid: B_*64 blocks of 256 threads; 4 pixels/thread (b128 loads/stores).
// x is 537 MB read-once -> non-temporal loads; out -> non-temporal stores.
// ---------------------------------------------------------------------------
__global__ __launch_bounds__(256) void torgb_main_kernel(
    const float* __restrict__ x,    // (16,128,256,256)
    const float* __restrict__ eff,  // ws: (16,3,128)
    const float* __restrict__ bias, // (3)
    float* __restrict__ out)        // (16,3,256,256)
{
  __shared__ float se[COUT_ * CIN_];

  const int b     = blockIdx.x >> 6;   // 64 pixel-chunks per batch image
  const int chunk = blockIdx.x & 63;

  for (int t = threadIdx.x; t < COUT_ * CIN_; t += 256)
    se[t] = eff[b * (COUT_ * CIN_) + t];
  __syncthreads();

  const int pix = chunk * 1024 + threadIdx.x * 4;
  const float* xp = x + ((size_t)b * CIN_) * HW_ + pix;

  f4 a0 = {0.f, 0.f, 0.f, 0.f};
  f4 a1 = a0, a2 = a0;

#pragma unroll 8
  for (int i = 0; i < CIN_; ++i) {
    const f4 xv = __builtin_nontemporal_load((const f4*)(xp + (size_t)i * HW_));
    const float e0 = se[i];
    const float e1 = se[CIN_ + i];
    const float e2 = se[2 * CIN_ + i];
    a0.x = fmaf(xv.x, e0, a0.x); a0.y = fmaf(xv.y, e0, a0.y);
    a0.z = fmaf(xv.z, e0, a0.z); a0.w = fmaf(xv.w, e0, a0.w);
    a1.x = fmaf(xv.x, e1, a1.x); a1.y = fmaf(xv.y, e1, a1.y);
    a1.z = fmaf(xv.z, e1, a1.z); a1.w = fmaf(xv.w, e1, a1.w);
    a2.x = fmaf(xv.x, e2, a2.x); a2.y = fmaf(xv.y, e2, a2.y);
    a2.z = fmaf(xv.z, e2, a2.z); a2.w = fmaf(xv.w, e2, a2.w);
  }

  const float b0 = bias[0], b1 = bias[1], b2 = bias[2];
  a0.x += b0; a0.y += b0; a0.z += b0; a0.w += b0;
  a1.x += b1; a1.y += b1; a1.z += b1; a1.w += b1;
  a2.x += b2; a2.y += b2; a2.z += b2; a2.w += b2;

  float* op = out + ((size_t)b * COUT_) * HW_ + pix;
  __builtin_nontemporal_store(a0, (f4*)(op + 0 * (size_t)HW_));
  __builtin_nontemporal_store(a1, (f4*)(op + 1 * (size_t)HW_));
  __builtin_nontemporal_store(a2, (f4*)(op + 2 * (size_t)HW_));
}

// ---------------------------------------------------------------------------
extern "C" void kernel_launch(void* const* d_in, const int* in_sizes, int n_in,
                              void* d_out, int out_size, void* d_ws, size_t ws_size,
                              hipStream_t stream) {
  (void)in_sizes; (void)n_in; (void)out_size; (void)ws_size;

  const float* x        = (const float*)d_in[0];
  const float* w        = (const float*)d_in[1];
  const float* weight   = (const float*)d_in[2];
  const float* bias     = (const float*)d_in[3];
  const float* affine_w = (const float*)d_in[4];
  const float* affine_b = (const float*)d_in[5];
  float*       out      = (float*)d_out;
  float*       eff      = (float*)d_ws;   // 16*3*128 floats = 24 KB

  style_eff_kernel<<<1, 256, 0, stream>>>(w, weight, affine_w, affine_b, eff);
  torgb_main_kernel<<<B_ * 64, 256, 0, stream>>>(x, eff, bias, out);
}